// GraphConvolution_33973191311763
// MI455X (gfx1250) — compile-verified
//
#include <hip/hip_runtime.h>
#include <math.h>

#define N_NODES 50000
#define FDIM    256
#define NEDGES  800000
#define PITCH   260          // LDS row pitch (floats): 260 % 64 == 4 -> bank-conflict free

typedef __attribute__((ext_vector_type(2))) float v2f;
typedef __attribute__((ext_vector_type(8))) float v8f;

__device__ __forceinline__ void atomic_add_f32(float* p, float v) {
    // relaxed, agent scope -> global_atomic_add_f32 on gfx1250
    __hip_atomic_fetch_add(p, v, __ATOMIC_RELAXED, __HIP_MEMORY_SCOPE_AGENT);
}

// deg[i] = 1.0 (self loop), agg[:] = 0  (agg aliases d_out)
__global__ void k_init(float* __restrict__ deg, float* __restrict__ agg) {
    int idx = blockIdx.x * blockDim.x + threadIdx.x;
    const int total4 = N_NODES * FDIM / 4;   // 3.2M float4
    if (idx < total4) {
        float4 z; z.x = 0.f; z.y = 0.f; z.z = 0.f; z.w = 0.f;
        ((float4*)agg)[idx] = z;
    }
    if (idx < N_NODES) deg[idx] = 1.0f;
}

// deg[dst[e]] += 1
__global__ void k_degree(const int* __restrict__ dst, float* __restrict__ deg) {
    int e = blockIdx.x * blockDim.x + threadIdx.x;
    if (e < NEDGES) atomic_add_f32(&deg[dst[e]], 1.0f);
}

__global__ void k_dinv(const float* __restrict__ deg, float* __restrict__ dinv) {
    int i = blockIdx.x * blockDim.x + threadIdx.x;
    if (i < N_NODES) dinv[i] = rsqrtf(deg[i]);
}

// one wave per edge: agg[src[e]][:] += H[dst[e]][:] * dinv[dst[e]]
__global__ void k_scatter(const float* __restrict__ H, const int* __restrict__ src,
                          const int* __restrict__ dst, const float* __restrict__ dinv,
                          float* __restrict__ agg) {
    int e = blockIdx.x * 8 + (threadIdx.x >> 5);
    if (e >= NEDGES) return;
    int lane = threadIdx.x & 31;
    int j = dst[e];
    int i = src[e];
    float dj = dinv[j];
    const float4* hrow = (const float4*)(H + (size_t)j * FDIM);
    float* arow = agg + (size_t)i * FDIM;
#pragma unroll
    for (int c = 0; c < 2; ++c) {
        int q = lane + 32 * c;            // float4 slot 0..63
        float4 v = hrow[q];
        atomic_add_f32(arow + q * 4 + 0, v.x * dj);
        atomic_add_f32(arow + q * 4 + 1, v.y * dj);
        atomic_add_f32(arow + q * 4 + 2, v.z * dj);
        atomic_add_f32(arow + q * 4 + 3, v.w * dj);
    }
}

// Fused: T = (1-a)*(dinv*agg + H*dinv^2) + a*H0 staged into LDS, then
// out = (1-beta)*T + beta*(T @ W) via v_wmma_f32_16x16x4_f32.
// Block = 512 threads = 16 waves; wave w computes column tile w; grid = N/16.
// AGG aliases d_out: each block reads only its own 16 rows (pre-barrier) and
// overwrites them (post-barrier) -> in-place safe.
__global__ __launch_bounds__(512) void k_gemm_fused(
        const float* __restrict__ H, const float* __restrict__ H0,
        float* AGG, const float* __restrict__ dinv,
        const float* __restrict__ W,
        const float* __restrict__ alpha_p, const float* __restrict__ lamda_p,
        const int* __restrict__ l_p) {
    __shared__ float Tlds[16 * PITCH];

    const int rowTile = blockIdx.x;            // 0..3124
    const int row0    = rowTile * 16;
    const int tid     = threadIdx.x;

    // ---- stage: compute 16x256 init_res tile into LDS ----
    const float alpha = *alpha_p;
    const float oma   = 1.0f - alpha;
#pragma unroll
    for (int s = tid; s < 1024; s += 512) {    // 1024 float4 slots
        int r  = s >> 6;                       // 0..15 local row
        int c4 = s & 63;                       // 0..63 float4 column
        size_t g4 = ((size_t)(row0 + r) * FDIM) / 4 + c4;
        float4 a4 = ((const float4*)AGG)[g4];
        float4 h4 = ((const float4*)H)[g4];
        float4 h0 = ((const float4*)H0)[g4];
        float  di = dinv[row0 + r];
        float  invdeg = di * di;
        float4 t;
        t.x = oma * (di * a4.x + h4.x * invdeg) + alpha * h0.x;
        t.y = oma * (di * a4.y + h4.y * invdeg) + alpha * h0.y;
        t.z = oma * (di * a4.z + h4.z * invdeg) + alpha * h0.z;
        t.w = oma * (di * a4.w + h4.w * invdeg) + alpha * h0.w;
        *(float4*)&Tlds[r * PITCH + c4 * 4] = t;
    }
    __syncthreads();

    // ---- WMMA: one 16x16 tile per wave ----
    const int wave = tid >> 5;                 // 0..15 = column tile
    const int lane = tid & 31;
    const int half = lane >> 4;                // 0/1
    const int l16  = lane & 15;
    const int ncol = wave * 16 + l16;          // B/C/D column
    const float* Wb = W + ncol;

    v8f acc = {};
#pragma unroll 4
    for (int k = 0; k < FDIM; k += 4) {
        int k0 = k + 2 * half;                 // lanes 16-31 take K+2,K+3
        v2f a, b;
        a.x = Tlds[l16 * PITCH + k0];
        a.y = Tlds[l16 * PITCH + k0 + 1];
        b.x = Wb[(size_t)k0 * FDIM];
        b.y = Wb[(size_t)(k0 + 1) * FDIM];
        acc = __builtin_amdgcn_wmma_f32_16x16x4_f32(
            /*neg_a=*/false, a, /*neg_b=*/false, b,
            /*c_mod=*/(short)0, acc, /*reuse_a=*/false, /*reuse_b=*/false);
    }

    const float beta = logf((*lamda_p) / (float)(*l_p) + 1.0f);
    const float omb  = 1.0f - beta;
#pragma unroll
    for (int r = 0; r < 8; ++r) {
        int rloc = r + 8 * half;               // C/D: lanes 16-31 hold rows M+8
        float t = Tlds[rloc * PITCH + ncol];
        AGG[(size_t)(row0 + rloc) * FDIM + ncol] = omb * t + beta * acc[r];
    }
}

extern "C" void kernel_launch(void* const* d_in, const int* in_sizes, int n_in,
                              void* d_out, int out_size, void* d_ws, size_t ws_size,
                              hipStream_t stream) {
    const float* H     = (const float*)d_in[0];
    const int*   edge  = (const int*)d_in[1];
    const float* H0    = (const float*)d_in[2];
    const float* W     = (const float*)d_in[3];
    const float* lamda = (const float*)d_in[4];
    const float* alpha = (const float*)d_in[5];
    const int*   l     = (const int*)d_in[6];
    float* out = (float*)d_out;

    const int* src = edge;            // edge_index[0]
    const int* dst = edge + NEDGES;   // edge_index[1]

    float* deg  = (float*)d_ws;                    // N floats
    float* dinv = deg + N_NODES;                   // N floats
    float* agg  = out;                             // agg aliases d_out

    const int total4 = N_NODES * FDIM / 4;         // 3,200,000
    k_init<<<(total4 + 255) / 256, 256, 0, stream>>>(deg, agg);
    k_degree<<<(NEDGES + 255) / 256, 256, 0, stream>>>(dst, deg);
    k_dinv<<<(N_NODES + 255) / 256, 256, 0, stream>>>(deg, dinv);
    k_scatter<<<NEDGES / 8, 256, 0, stream>>>(H, src, dst, dinv, agg);
    k_gemm_fused<<<N_NODES / 16, 512, 0, stream>>>(H, H0, agg, dinv, W,
                                                   alpha, lamda, l);
}